// YoloV1Loss_21930103014188
// MI455X (gfx1250) — compile-verified
//
#include <hip/hip_runtime.h>
#include <cstdint>
#include <cstddef>

// ---------------- problem constants (from reference) ----------------
#define SS 7
#define NB 2              // boxes per cell
#define NC 20             // classes
#define CH 30             // NB*5 + NC channels per cell
#define CELLS 49          // SS*SS
#define TT 32             // targets per image (== wave32 lanes)
#define FLOATS_PER_IMG 1470   // CELLS*CH
#define QWORDS_PER_IMG 735    // FLOATS_PER_IMG/2 (image slab is 8B aligned)
#define IMG_PER_BLOCK 8       // 8 waves / 256-thread block
#define LDS_STRIDE 1472       // padded per-image stride (floats), 8B aligned

// ---------------- CDNA5 async global->LDS path ----------------
#if defined(__has_builtin)
#if __has_builtin(__builtin_amdgcn_global_load_async_to_lds_b64)
#define HAVE_ASYNC_LDS 1
#endif
#endif

typedef int v2i_t __attribute__((ext_vector_type(2)));
typedef __attribute__((address_space(1))) v2i_t gas1_v2i;
typedef __attribute__((address_space(3))) v2i_t gas3_v2i;

__device__ __forceinline__ void stage_qword(const float* g, float* l) {
#ifdef HAVE_ASYNC_LDS
  // GLOBAL_LOAD_ASYNC_TO_LDS_B64: tracked on ASYNCcnt, no VGPR data
  __builtin_amdgcn_global_load_async_to_lds_b64((gas1_v2i*)g, (gas3_v2i*)l,
                                                /*offset=*/0, /*cpol=*/0);
#else
  *(float2*)l = *(const float2*)g;
#endif
}

__device__ __forceinline__ void wait_async_done() {
#ifdef HAVE_ASYNC_LDS
#if __has_builtin(__builtin_amdgcn_s_wait_asynccnt)
  __builtin_amdgcn_s_wait_asynccnt(0);
#else
  asm volatile("s_wait_asynccnt 0" ::: "memory");
#endif
  asm volatile("" ::: "memory");
#endif
}

__device__ __forceinline__ float wave32_sum(float v) {
#pragma unroll
  for (int m = 16; m > 0; m >>= 1) v += __shfl_xor(v, m, 32);
  return v;
}

// ---------------- main kernel: one wave per image ----------------
__global__ __launch_bounds__(256) void yolo_v1_loss_kernel(
    const float* __restrict__ outputs,   // [N,7,7,30]
    const float* __restrict__ tboxes,    // [N,32,4]
    const int*   __restrict__ tlabels,   // [N,32]
    float* __restrict__ part,            // per-image partial (nullptr -> atomic path)
    float* __restrict__ out,             // scalar accumulator (atomic path only)
    int N) {
  __shared__ __align__(16) float slab[IMG_PER_BLOCK * LDS_STRIDE];

  const int lane = threadIdx.x & 31;
  const int wave = threadIdx.x >> 5;
  const int n    = blockIdx.x * IMG_PER_BLOCK + wave;
  const bool active = (n < N);
  float* lbase = slab + wave * LDS_STRIDE;

  if (active) {
    const float* gsrc = outputs + (size_t)n * FLOATS_PER_IMG;
    for (int i = lane; i < QWORDS_PER_IMG; i += 32)
      stage_qword(gsrc + 2 * i, lbase + 2 * i);
    wait_async_done();  // per-wave: ASYNCcnt covers this wave's own loads
  }
  __syncthreads();
  if (!active) return;

  const float cell = 1.0f / 7.0f;

  // ---- per-target (lane == target index, T == 32) ----
  const float4 tb = ((const float4*)tboxes)[(size_t)n * TT + lane];
  const int label = tlabels[(size_t)n * TT + lane];
  const float tx = tb.x, ty = tb.y, tw = tb.z, th = tb.w;

  int gxi = (int)(tx / cell); gxi = gxi < 0 ? 0 : (gxi > 6 ? 6 : gxi);
  int gyi = (int)(ty / cell); gyi = gyi < 0 ? 0 : (gyi > 6 ? 6 : gyi);
  const float ox = (tx - (float)gxi * cell) / cell;
  const float oy = (ty - (float)gyi * cell) / cell;

  const float* cb = lbase + (gyi * SS + gxi) * CH;  // this target's cell

  float iou[2], px[2], py[2], pw[2], ph[2];
#pragma unroll
  for (int b = 0; b < 2; ++b) {
    const float bx = cb[4 * b + 0], by = cb[4 * b + 1];
    pw[b] = cb[4 * b + 2];
    ph[b] = cb[4 * b + 3];
    px[b] = (bx + (float)gxi) * cell;
    py[b] = (by + (float)gyi) * cell;
    const float x11 = tx - tw * 0.5f, x12 = tx + tw * 0.5f;
    const float y11 = ty - th * 0.5f, y12 = ty + th * 0.5f;
    const float x21 = px[b] - pw[b] * 0.5f, x22 = px[b] + pw[b] * 0.5f;
    const float y21 = py[b] - ph[b] * 0.5f, y22 = py[b] + ph[b] * 0.5f;
    const float iw = fmaxf(fminf(x12, x22) - fmaxf(x11, x21), 0.0f);
    const float ih = fmaxf(fminf(y12, y22) - fmaxf(y11, y21), 0.0f);
    const float inter = iw * ih;
    const float uni = tw * th + pw[b] * ph[b] - inter;
    iou[b] = inter / fmaxf(uni, 1e-6f);
  }
  const int best = (iou[1] > iou[0]) ? 1 : 0;  // argmax, first-on-tie

  const float bpw = fmaxf(pw[best], 1e-6f), bph = fmaxf(ph[best], 1e-6f);
  const float twc = fmaxf(tw, 1e-6f),       thc = fmaxf(th, 1e-6f);
  const float dx = px[best] - ox, dy = py[best] - oy;
  const float dw = sqrtf(bpw) - sqrtf(twc);
  const float dh = sqrtf(bph) - sqrtf(thc);
  const float box_loss = dx * dx + dy * dy + dw * dw + dh * dh;

  const float dconf = cb[8 + best] - 1.0f;
  const float obj_loss = dconf * dconf;

  // ---- class loss: mean over C of (softmax - onehot)^2 ----
  float cv[NC];
  float mx = cb[10];
#pragma unroll
  for (int k = 1; k < NC; ++k) mx = fmaxf(mx, cb[10 + k]);
  float ssum = 0.0f;
#pragma unroll
  for (int k = 0; k < NC; ++k) { cv[k] = expf(cb[10 + k] - mx); ssum += cv[k]; }
  const float inv = 1.0f / ssum;
  float cls_loss = 0.0f;
#pragma unroll
  for (int k = 0; k < NC; ++k) {
    const float d = cv[k] * inv - ((k == label) ? 1.0f : 0.0f);
    cls_loss += d * d;
  }
  cls_loss *= (1.0f / (float)NC);

  // ---- no-object: mean over 98 conf values of conf^2 (per image) ----
  float nb = 0.0f;
  for (int c = lane; c < CELLS; c += 32) {
    const float c0 = lbase[c * CH + 8];
    const float c1 = lbase[c * CH + 9];
    nb += c0 * c0 + c1 * c1;
  }

  float total = 5.0f * box_loss + obj_loss + cls_loss + nb * (0.5f / 98.0f);
  total = wave32_sum(total);

  if (lane == 0) {
    if (part) part[n] = total;
    else      atomicAdd(out, total / (float)N);
  }
}

// ---------------- deterministic final reduction ----------------
__global__ __launch_bounds__(256) void yolo_reduce_kernel(
    const float* __restrict__ part, float* __restrict__ out, int N) {
  __shared__ float sm[256];
  float s = 0.0f;
  for (int i = threadIdx.x; i < N; i += 256) s += part[i];
  sm[threadIdx.x] = s;
  __syncthreads();
#pragma unroll
  for (int off = 128; off > 0; off >>= 1) {
    if ((int)threadIdx.x < off) sm[threadIdx.x] += sm[threadIdx.x + off];
    __syncthreads();
  }
  if (threadIdx.x == 0) out[0] = sm[0] / (float)N;
}

__global__ void yolo_zero_kernel(float* out) { out[0] = 0.0f; }

// ---------------- harness entry ----------------
extern "C" void kernel_launch(void* const* d_in, const int* in_sizes, int n_in,
                              void* d_out, int out_size, void* d_ws, size_t ws_size,
                              hipStream_t stream) {
  (void)n_in; (void)out_size;
  const float* outputs = (const float*)d_in[0];
  const float* tboxes  = (const float*)d_in[1];
  const int*   tlabels = (const int*)d_in[2];
  float* out = (float*)d_out;

  const int N = in_sizes[1] / (TT * 4);       // target_boxes is [N,32,4]
  const int blocks = (N + IMG_PER_BLOCK - 1) / IMG_PER_BLOCK;

  const bool two_pass = (ws_size >= (size_t)N * sizeof(float));
  float* part = two_pass ? (float*)d_ws : nullptr;

  if (!two_pass)
    yolo_zero_kernel<<<1, 1, 0, stream>>>(out);

  yolo_v1_loss_kernel<<<blocks, 256, 0, stream>>>(outputs, tboxes, tlabels,
                                                  part, out, N);
  if (two_pass)
    yolo_reduce_kernel<<<1, 256, 0, stream>>>(part, out, N);
}